// Subtraction2_83614423319318
// MI455X (gfx1250) — compile-verified
//
#include <hip/hip_runtime.h>
#include <hip/hip_bf16.h>

// Problem constants (from reference): N=8, C=32, H=W=112, K=7, pad=3, stride=1
// Output: [8, 32, 49, 12544] f32 = 629.4 MB  -> pure streaming-store bound.
#define PLANES   256          // N*C
#define HW       112
#define LSZ      12544        // 112*112
#define TILE     16           // rows per block
#define NTILES   7            // 112 / 16
#define LDSW     128          // padded LDS row width (floats), data at col x+4
#define LDSROWS  22           // TILE + 6 halo rows
#define NTHREADS 448          // 16 rows * 28 float4-chunks

typedef float f4 __attribute__((ext_vector_type(4)));
typedef int   v4i __attribute__((ext_vector_type(4)));

#if defined(__has_builtin)
#if __has_builtin(__builtin_amdgcn_global_load_async_to_lds_b128) && \
    __has_builtin(__builtin_amdgcn_s_wait_asynccnt)
#define USE_ASYNC_LDS 1
#endif
#endif
#ifndef USE_ASYNC_LDS
#define USE_ASYNC_LDS 0
#endif

__global__ __launch_bounds__(NTHREADS)
void sub2_kernel(const float* __restrict__ in1,
                 const float* __restrict__ in2,
                 float* __restrict__ out) {
    __shared__ float s2[LDSROWS * LDSW];

    const int t     = threadIdx.x;        // 0..447
    const int plane = blockIdx.x;         // 0..255  (n*C + c)
    const int y0    = blockIdx.y * TILE;  // tile base row

    const float* __restrict__ p2 = in2 + (size_t)plane * LSZ;

    // ---- zero-fill LDS tile (provides the zero padding halo for free) ----
    {
        f4 z = (f4)0.0f;
        f4* s2v = (f4*)s2;
        for (int i = t; i < (LDSROWS * LDSW) / 4; i += NTHREADS) s2v[i] = z;
    }
    __syncthreads();

    // ---- stage valid in2 rows [y0-3, y0+18] into LDS, +4-col shift keeps
    //      every b128 LDS write 16B aligned ----
    for (int id = t; id < LDSROWS * 28; id += NTHREADS) {
        const int r  = id / 28;           // LDS row 0..21
        const int xc = id % 28;           // float4 chunk 0..27
        const int gy = y0 - 3 + r;        // global row
        if (gy >= 0 && gy < HW) {
            const float* g = p2 + gy * HW + xc * 4;
            float*       l = s2 + r * LDSW + 4 + xc * 4;
#if USE_ASYNC_LDS
            __builtin_amdgcn_global_load_async_to_lds_b128(
                (__attribute__((address_space(1))) v4i*)g,
                (__attribute__((address_space(3))) v4i*)l,
                0, 0);
#else
            *(f4*)l = *(const f4*)g;
#endif
        }
    }
#if USE_ASYNC_LDS
    __builtin_amdgcn_s_wait_asynccnt(0);
#endif
    __syncthreads();

    // ---- each thread owns one (row, float4 column chunk) of the output ----
    const int yl = t / 28;                // 0..15
    const int xc = t % 28;                // 0..27
    const int y  = y0 + yl;               // global output row

    const float* __restrict__ p1 = in1 + (size_t)plane * LSZ;
    const f4 center = *(const f4*)(p1 + y * HW + xc * 4);

    float* __restrict__ outp =
        out + (size_t)plane * 49 * LSZ + (size_t)y * HW + (size_t)xc * 4;

#pragma unroll
    for (int i = 0; i < 7; ++i) {
        // aligned 12-float window covering LDS cols [4*xc .. 4*xc+11];
        // needed neighbor cols are [4*xc+1 .. 4*xc+10]
        const float* row = s2 + (yl + i) * LDSW + xc * 4;
        float w[12];
        *(f4*)(&w[0]) = *(const f4*)(row);
        *(f4*)(&w[4]) = *(const f4*)(row + 4);
        *(f4*)(&w[8]) = *(const f4*)(row + 8);
#pragma unroll
        for (int j = 0; j < 7; ++j) {
            f4 v;
            v[0] = center[0] - w[j + 1];
            v[1] = center[1] - w[j + 2];
            v[2] = center[2] - w[j + 3];
            v[3] = center[3] - w[j + 4];
            // 629 MB output stream (>3x the 192MB L2): bypass with NT hint
            __builtin_nontemporal_store(
                v, (f4*)(outp + (size_t)(i * 7 + j) * LSZ));
        }
    }
}

extern "C" void kernel_launch(void* const* d_in, const int* in_sizes, int n_in,
                              void* d_out, int out_size, void* d_ws, size_t ws_size,
                              hipStream_t stream) {
    const float* in1 = (const float*)d_in[0];
    const float* in2 = (const float*)d_in[1];
    float* out = (float*)d_out;
    dim3 grid(PLANES, NTILES);
    sub2_kernel<<<grid, dim3(NTHREADS), 0, stream>>>(in1, in2, out);
}